// Adapt_SIMLoss_75591424409827
// MI455X (gfx1250) — compile-verified
//
#include <hip/hip_runtime.h>
#include <math.h>

typedef __attribute__((ext_vector_type(2))) float v2f;
typedef __attribute__((ext_vector_type(8))) float v8f;

// Problem constants (fixed by the reference): N=4, C=3, H=W=64 -> HW=4096.
static constexpr int kN  = 4;
static constexpr int kC  = 3;
static constexpr int kHW = 4096;

// abs+accumulate in ONE VALU op: v_add_f32_e64 acc, |x|, acc
#define ABS_ACC(A, X) \
  asm("v_add_f32_e64 %0, |%1|, %0" : "+v"(A) : "v"(X))

// ---------------------------------------------------------------------------
// Kernel 1: gating MLP + softmax + fuse, emit D = s_gt - fuse_fea in padded
// [n][pixel][4] float4 records (K-pad with 0), and repack gt0/gt1 the same
// way so WMMA A- and B-tiles are a single aligned b64 load per lane.
// ---------------------------------------------------------------------------
__global__ void __launch_bounds__(256) prep_kernel(
    const float* __restrict__ gt0, const float* __restrict__ gt1,
    const float* __restrict__ s_gt, const float* __restrict__ t_gt,
    const float* __restrict__ t_gtout,
    const float* __restrict__ w1, const float* __restrict__ b1,
    const float* __restrict__ w2, const float* __restrict__ b2,
    float* __restrict__ dbuf, float* __restrict__ gtp) {
  const int idx = blockIdx.x * blockDim.x + threadIdx.x;   // 0 .. 16383
  const int n = idx >> 12;
  const int i = idx & (kHW - 1);
  const size_t base = (size_t)n * kC * kHW + i;

  float f6[6];
#pragma unroll
  for (int c = 0; c < kC; ++c) {
    f6[c]     = t_gt   [base + (size_t)c * kHW];
    f6[3 + c] = t_gtout[base + (size_t)c * kHW];
  }

  // Conv2d(6,12,1) -> ReLU -> Conv2d(12,2,1)
  float l0 = b2[0], l1 = b2[1];
#pragma unroll
  for (int o = 0; o < 12; ++o) {
    float h = b1[o];
#pragma unroll
    for (int c = 0; c < 6; ++c) h = fmaf(w1[o * 6 + c], f6[c], h);
    h = fmaxf(h, 0.0f);
    l0 = fmaf(w2[o],      h, l0);
    l1 = fmaf(w2[12 + o], h, l1);
  }
  // softmax over the 2 channels
  const float mx = fmaxf(l0, l1);
  const float e0 = __expf(l0 - mx), e1 = __expf(l1 - mx);
  const float inv = 1.0f / (e0 + e1);
  const float s0 = e0 * inv, s1 = e1 * inv;

  float* dp = dbuf + (size_t)idx * 4;
#pragma unroll
  for (int k = 0; k < kC; ++k) {
    const float fuse = f6[k] * s0 + f6[3 + k] * s1;
    dp[k] = s_gt[base + (size_t)k * kHW] - fuse;
  }
  dp[3] = 0.0f;

  float* g0 = gtp + (((size_t)(0 * kN + n) * kHW) + i) * 4;
  float* g1 = gtp + (((size_t)(1 * kN + n) * kHW) + i) * 4;
#pragma unroll
  for (int k = 0; k < kC; ++k) {
    g0[k] = gt0[base + (size_t)k * kHW];
    g1[k] = gt1[base + (size_t)k * kHW];
  }
  g0[3] = 0.0f;
  g1[3] = 0.0f;
}

// ---------------------------------------------------------------------------
// Kernel 2: rank-4 (K padded 3->4) tiled similarity-difference reduction.
// One wave32 owns TWO adjacent 16-row A-tiles of one (gti, n) similarity
// matrix and sweeps all 256 B column-tiles: 2 x v_wmma_f32_16x16x4_f32 per
// global_load_b64; |D| accumulated with single-op v_add_f32 |src|.
// 1024 waves total.
// ---------------------------------------------------------------------------
__global__ void __launch_bounds__(256) simloss_wmma(
    const float* __restrict__ dbuf, const float* __restrict__ gtp,
    float* __restrict__ out) {
  const int lane = threadIdx.x & 31;
  const int wid  = (blockIdx.x * blockDim.x + threadIdx.x) >> 5;  // 0 .. 1023
  const int gti  = wid >> 9;           // 0 -> gt0 (w=0.02), 1 -> gt1 (w=1.0)
  const int n    = (wid >> 7) & 3;
  const int pblk = (wid & 127) << 1;   // first of two consecutive row-tiles

  // ISA layout for 32-bit 16x4 A (and mirrored 4x16 B):
  //   lanes 0-15: M(or N)=lane, VGPR0=K0, VGPR1=K1
  //   lanes16-31: M(or N)=lane-16, VGPR0=K2, VGPR1=K3
  const int m  = lane & 15;
  const int kb = (lane >> 4) << 1;     // 0 or 2; [k][k+1] is one aligned b64

  const float* aP = dbuf + (((size_t)n * kHW) + pblk * 16 + m) * 4 + kb;
  const v2f a0 = *(const v2f*)aP;
  const v2f a1 = *(const v2f*)(aP + 16 * 4);   // next 16-row tile

  const float* bBase =
      gtp + ((size_t)(gti * kN + n) * kHW) * 4 + (size_t)m * 4 + kb;

  // 16 scalar accumulators (kept in VGPRs; tied "+v" asm operands)
  float s00 = 0.f, s01 = 0.f, s02 = 0.f, s03 = 0.f;
  float s04 = 0.f, s05 = 0.f, s06 = 0.f, s07 = 0.f;
  float s10 = 0.f, s11 = 0.f, s12 = 0.f, s13 = 0.f;
  float s14 = 0.f, s15 = 0.f, s16 = 0.f, s17 = 0.f;

#pragma unroll 4
  for (int q = 0; q < kHW; q += 16) {
    const v2f b = *(const v2f*)(bBase + (size_t)q * 4);
    v8f c0 = {}, c1 = {};
    // (neg_a, A, neg_b, B, c_mod, C, reuse_a, reuse_b)
    c0 = __builtin_amdgcn_wmma_f32_16x16x4_f32(
        false, a0, false, b, (short)0, c0, false, false);
    c1 = __builtin_amdgcn_wmma_f32_16x16x4_f32(
        false, a1, false, b, (short)0, c1, false, false);
    ABS_ACC(s00, c0[0]); ABS_ACC(s01, c0[1]);
    ABS_ACC(s02, c0[2]); ABS_ACC(s03, c0[3]);
    ABS_ACC(s04, c0[4]); ABS_ACC(s05, c0[5]);
    ABS_ACC(s06, c0[6]); ABS_ACC(s07, c0[7]);
    ABS_ACC(s10, c1[0]); ABS_ACC(s11, c1[1]);
    ABS_ACC(s12, c1[2]); ABS_ACC(s13, c1[3]);
    ABS_ACC(s14, c1[4]); ABS_ACC(s15, c1[5]);
    ABS_ACC(s16, c1[6]); ABS_ACC(s17, c1[7]);
  }

  float v = ((s00 + s01) + (s02 + s03)) + ((s04 + s05) + (s06 + s07)) +
            ((s10 + s11) + (s12 + s13)) + ((s14 + s15) + (s16 + s17));
  // wave32 tree reduction
#pragma unroll
  for (int off = 16; off > 0; off >>= 1) v += __shfl_xor(v, off, 32);

  if (lane == 0) {
    const float w = (gti == 0) ? 0.02f : 1.0f;
    // mean over N*HW*HW = 4*4096*4096 elements
    atomicAdd(out, v * (w / 67108864.0f));
  }
}

// ---------------------------------------------------------------------------
extern "C" void kernel_launch(void* const* d_in, const int* in_sizes, int n_in,
                              void* d_out, int out_size, void* d_ws,
                              size_t ws_size, hipStream_t stream) {
  (void)in_sizes; (void)n_in; (void)out_size; (void)ws_size;
  const float* gt0     = (const float*)d_in[0];
  const float* gt1     = (const float*)d_in[1];
  const float* s_gt    = (const float*)d_in[2];
  const float* t_gt    = (const float*)d_in[3];
  const float* t_gtout = (const float*)d_in[4];
  const float* w1      = (const float*)d_in[5];
  const float* b1      = (const float*)d_in[6];
  const float* w2      = (const float*)d_in[7];
  const float* b2      = (const float*)d_in[8];

  float* dbuf = (float*)d_ws;                       // N*HW*4 floats = 256 KB
  float* gtp  = dbuf + (size_t)kN * kHW * 4;        // 2*N*HW*4 floats = 512 KB
  float* out  = (float*)d_out;

  hipMemsetAsync(out, 0, sizeof(float), stream);
  prep_kernel<<<(kN * kHW) / 256, 256, 0, stream>>>(
      gt0, gt1, s_gt, t_gt, t_gtout, w1, b1, w2, b2, dbuf, gtp);
  simloss_wmma<<<128, 256, 0, stream>>>(dbuf, gtp, out);
}